// DynamicPatching_15994458211135
// MI455X (gfx1250) — compile-verified
//
#include <hip/hip_runtime.h>
#include <hip/hip_bf16.h>
#include <stdint.h>

// Problem constants from the reference: B=32, C=64, T=8192, S=64.
#define BB 32
#define CC 64
#define TT 8192
#define SS 64

// ---------------------------------------------------------------------------
// Device-side TDM availability (device pass only; host launch path is
// identical either way — single kernel name, body switched).
// ---------------------------------------------------------------------------
#if defined(__AMDGCN__) && defined(__has_builtin)
#if __has_builtin(__builtin_amdgcn_tensor_load_to_lds) && \
    __has_builtin(__builtin_amdgcn_tensor_store_from_lds) && \
    __has_builtin(__builtin_amdgcn_s_wait_tensorcnt)
#define USE_TDM 1
#endif
#endif

typedef unsigned int u32x4 __attribute__((ext_vector_type(4)));
typedef int i32x4 __attribute__((ext_vector_type(4)));
typedef int i32x8 __attribute__((ext_vector_type(8)));

#ifdef USE_TDM
// D# group 0: [1:0]=count=1, [63:32]=lds_addr, [120:64]=global_addr,
// [127:126]=type=2 ("image").
__device__ __forceinline__ u32x4 tdm_g0(uint32_t lds_addr, uint64_t gaddr) {
  u32x4 g;
  g.x = 1u;                                            // count=1, user mode
  g.y = lds_addr;                                      // LDS byte address
  g.z = (uint32_t)(gaddr & 0xFFFFFFFFu);               // global_addr[31:0]
  g.w = (uint32_t)((gaddr >> 32) & 0x1FFFFFFu)         // global_addr[56:32]
        | (2u << 30);                                  // type=2
  return g;
}

// D# group 1 for a 2D tile of 4-byte elements:
//   [17:16]=data_size=2 (4B), [79:48]=tensor_dim0, [111:80]=tensor_dim1,
//   [127:112]=tile_dim0, [143:128]=tile_dim1, [159:144]=tile_dim2=0,
//   [207:160]=tensor_dim0_stride, [255:208]=tensor_dim1_stride=0.
__device__ __forceinline__ i32x8 tdm_g1_2d(uint32_t tdim0, uint32_t tdim1,
                                           uint32_t tile0, uint32_t tile1,
                                           uint64_t stride0) {
  i32x8 g;
  g[0] = (int)(2u << 16);                                             // 4B elems
  g[1] = (int)((tdim0 & 0xFFFFu) << 16);                              // tdim0 lo
  g[2] = (int)(((tdim0 >> 16) & 0xFFFFu) | ((tdim1 & 0xFFFFu) << 16));
  g[3] = (int)(((tdim1 >> 16) & 0xFFFFu) | ((tile0 & 0xFFFFu) << 16));
  g[4] = (int)(tile1 & 0xFFFFu);                                      // tile2=0
  g[5] = (int)(uint32_t)(stride0 & 0xFFFFFFFFu);
  g[6] = (int)(uint32_t)((stride0 >> 32) & 0xFFFFu);                  // d1s=0
  g[7] = 0;
  return g;
}
#endif

// One single-wave workgroup per (b, s) patch; blockIdx.x = b*S + s.
// The wave issues one TDM load (global tile -> LDS, with DMA-side zero-fill
// implementing the segment mask) and one TDM store (LDS -> padded output
// tile), then retires. All data movement is done by the Tensor Data Mover.
__global__ void dynamic_patch_kernel(const float* __restrict__ tensor,
                                     const int* __restrict__ cps,
                                     float* __restrict__ out,
                                     int max_len) {
  const int bs = (int)blockIdx.x;          // 0 .. B*S-1
  const int b  = bs >> 6;                  // / S (S == 64)
  const int s  = bs & (SS - 1);            // % S

#ifdef USE_TDM
  extern __shared__ float lds_tile[];      // C * max_len floats (dynamic)
  (void)lds_tile;

  int start = __builtin_amdgcn_readfirstlane(cps[b * (SS + 1) + s]);
  int end   = __builtin_amdgcn_readfirstlane(cps[b * (SS + 1) + s + 1]);
  int L     = end - start;
  if (L < 0) L = 0;

  uint32_t lds_off = 0u;
#if __has_builtin(__builtin_amdgcn_groupstaticsize)
  lds_off = (uint32_t)__builtin_amdgcn_groupstaticsize();
#endif

  // Source tile start: &tensor[b, 0, start]; rows strided by T.
  const uint64_t src = (uint64_t)(uintptr_t)tensor +
                       4ull * ((uint64_t)b * CC * TT + (uint64_t)start);
  // Dest tile start: &out[b, s, 0, 0]; rows strided by max_len (contiguous).
  const uint64_t dst = (uint64_t)(uintptr_t)out +
                       4ull * ((uint64_t)bs * CC * (uint64_t)max_len);

  // LOAD: tile = max_len x C rows; tensor_dim0 = L so X >= L reads return
  // zero -> the reference mask is implemented by the DMA engine.
  u32x4 ld_g0 = tdm_g0(lds_off, src);
  i32x8 ld_g1 = tdm_g1_2d((uint32_t)L, (uint32_t)CC,
                          (uint32_t)max_len, (uint32_t)CC,
                          (uint64_t)TT);
  // STORE: same tile shape, dense rows of length max_len into out
  // (writes the zero padding too — d_out is poisoned by the harness).
  u32x4 st_g0 = tdm_g0(lds_off, dst);
  i32x8 st_g1 = tdm_g1_2d((uint32_t)max_len, (uint32_t)CC,
                          (uint32_t)max_len, (uint32_t)CC,
                          (uint64_t)max_len);

  i32x4 zero4 = {0, 0, 0, 0};
#if __clang_major__ >= 23
  i32x8 zero8 = {0, 0, 0, 0, 0, 0, 0, 0};
  __builtin_amdgcn_tensor_load_to_lds(ld_g0, ld_g1, zero4, zero4, zero8, 0);
  __builtin_amdgcn_s_wait_tensorcnt(0);
  __builtin_amdgcn_tensor_store_from_lds(st_g0, st_g1, zero4, zero4, zero8, 0);
#else
  __builtin_amdgcn_tensor_load_to_lds(ld_g0, ld_g1, zero4, zero4, 0);
  __builtin_amdgcn_s_wait_tensorcnt(0);
  __builtin_amdgcn_tensor_store_from_lds(st_g0, st_g1, zero4, zero4, 0);
#endif
  // S_ENDPGM implies wait-idle, but make the dependency explicit.
  __builtin_amdgcn_s_wait_tensorcnt(0);
#else
  // Fallback: plain coalesced masked copy (also what the host stub parses).
  const int start = cps[b * (SS + 1) + s];
  const int end   = cps[b * (SS + 1) + s + 1];
  const int L     = end - start;
  const float* src = tensor + (size_t)b * CC * TT;
  float* dst       = out + (size_t)bs * CC * (size_t)max_len;
  const int n = CC * max_len;
  for (int i = (int)threadIdx.x; i < n; i += (int)blockDim.x) {
    const int c = i / max_len;
    const int p = i - c * max_len;
    float v = 0.0f;
    if (p < L) v = src[c * TT + start + p];
    dst[i] = v;
  }
#endif
}

extern "C" void kernel_launch(void* const* d_in, const int* in_sizes, int n_in,
                              void* d_out, int out_size, void* d_ws, size_t ws_size,
                              hipStream_t stream) {
  (void)in_sizes; (void)n_in; (void)d_ws; (void)ws_size;
  const float* tensor = (const float*)d_in[0];     // (B, C, T) f32
  const int*   cps    = (const int*)d_in[1];       // (B, S+1) int32
  float*       out    = (float*)d_out;             // (B, S, C, max_len) f32

  // Derive max_len from the output size (avoids reading the device scalar).
  const int max_len = out_size / (BB * SS * CC);
  if (max_len <= 0) return;

  const size_t shmem = (size_t)CC * (size_t)max_len * sizeof(float); // ~56 KB
  dim3 grid(BB * SS);   // 2048 patches, one DMA pair each
  dim3 block(32);       // exactly one wave drives the TDM per patch
  dynamic_patch_kernel<<<grid, block, shmem, stream>>>(tensor, cps, out, max_len);
}